// IF_TBN_Torch_53541062312290
// MI455X (gfx1250) — compile-verified
//
#include <hip/hip_runtime.h>
#include <cstdint>

// Integrate-and-fire scan over time, [T=32, B, C, H, W] fp32.
// Memory-bound streaming scan: state (mem) lives in VGPRs, input is staged
// through LDS with the CDNA5 async global->LDS engine (ASYNCcnt pipeline),
// output spikes are streamed out with non-temporal 128-bit stores.

typedef float f4 __attribute__((ext_vector_type(4)));

#define TSTEPS 32
#define BATCH  4      // async loads in flight per issue group
#define NBUF   2      // double buffer of BATCH slots
#define BLK    256    // threads per block (8 wave32)
#define V_TH   1.0f

// One async 16-byte-per-lane global->LDS copy (GVS mode: sgpr base + vgpr offset).
#define ASYNC_LOAD16(ldsaddr, goff, base64)                                 \
  asm volatile("global_load_async_to_lds_b128 %0, %1, %2"                   \
               :: "v"(ldsaddr), "v"(goff), "s"(base64) : "memory")

__global__ __launch_bounds__(BLK) void if_scan_async(const float* __restrict__ x,
                                                     float* __restrict__ out,
                                                     unsigned n_per_t) {
  // NBUF * BATCH slots, each slot = BLK lanes * 16B = 4 KiB  -> 32 KiB LDS
  __shared__ __align__(16) float lbuf[NBUF * BATCH * BLK * 4];

  const unsigned tid = threadIdx.x;
  const unsigned long long blkBase = (unsigned long long)blockIdx.x * (BLK * 4);
  if (blkBase >= n_per_t) return;                      // uniform guard

  const unsigned e0         = (unsigned)blkBase + tid * 4;  // element idx in slice
  const bool     active     = (e0 + 3u) < n_per_t;          // per-lane tail guard
  const unsigned laneByte   = e0 * 4u;                      // byte offset in slice
  const unsigned sliceBytes = n_per_t * 4u;                 // 16 MiB per t (fits u32*T)

  const unsigned ldsBase   = (unsigned)(uintptr_t)(&lbuf[0]); // LDS byte offset
  const unsigned slotBytes = BLK * 16u;                       // 4096
  const unsigned myLds     = ldsBase + tid * 16u;

  const uint64_t xb = (uint64_t)(uintptr_t)x;

  // ---- prologue: issue batch 0 into buffer 0 ----
  if (active) {
#pragma unroll
    for (int k = 0; k < BATCH; ++k) {
      unsigned goff = (unsigned)k * sliceBytes + laneByte;
      unsigned la   = myLds + (unsigned)k * slotBytes;
      ASYNC_LOAD16(la, goff, xb);
    }
  }

  float m0 = 0.0f, m1 = 0.0f, m2 = 0.0f, m3 = 0.0f;

#pragma unroll
  for (int b = 0; b < TSTEPS / BATCH; ++b) {
    const int cur = b & 1;

    if (b + 1 < TSTEPS / BATCH) {
      // issue next batch into the other buffer, then wait for current batch
      const int nxt = (b + 1) & 1;
      if (active) {
#pragma unroll
        for (int k = 0; k < BATCH; ++k) {
          unsigned t    = (unsigned)(b + 1) * BATCH + (unsigned)k;
          unsigned goff = t * sliceBytes + laneByte;
          unsigned la   = myLds + ((unsigned)nxt * BATCH + (unsigned)k) * slotBytes;
          ASYNC_LOAD16(la, goff, xb);
        }
      }
      // <= BATCH outstanding means the older batch (in-order) fully landed
      asm volatile("s_wait_asynccnt 0x4" ::: "memory");
    } else {
      asm volatile("s_wait_asynccnt 0x0" ::: "memory");
    }

    if (active) {
#pragma unroll
      for (int k = 0; k < BATCH; ++k) {
        const unsigned t = (unsigned)b * BATCH + (unsigned)k;
        const f4* lp = (const f4*)&lbuf[((size_t)cur * BATCH + k) * (BLK * 4) + tid * 4];
        f4 v = *lp;                                   // ds_load_b128

        f4 sp;
        m0 += v.x; sp.x = (m0 >= V_TH) ? 1.0f : 0.0f; m0 = (m0 >= V_TH) ? 0.0f : m0;
        m1 += v.y; sp.y = (m1 >= V_TH) ? 1.0f : 0.0f; m1 = (m1 >= V_TH) ? 0.0f : m1;
        m2 += v.z; sp.z = (m2 >= V_TH) ? 1.0f : 0.0f; m2 = (m2 >= V_TH) ? 0.0f : m2;
        m3 += v.w; sp.w = (m3 >= V_TH) ? 1.0f : 0.0f; m3 = (m3 >= V_TH) ? 0.0f : m3;

        unsigned long long oo = (unsigned long long)t * n_per_t + e0;
        __builtin_nontemporal_store(sp, (f4*)(out + oo));  // global_store_b128 NT
      }
    }
  }
}

extern "C" void kernel_launch(void* const* d_in, const int* in_sizes, int n_in,
                              void* d_out, int out_size, void* d_ws, size_t ws_size,
                              hipStream_t stream) {
  (void)n_in; (void)d_ws; (void)ws_size; (void)out_size;
  const float* x  = (const float*)d_in[0];
  float*      out = (float*)d_out;

  const long long total  = (long long)in_sizes[0];       // T*B*C*H*W
  const unsigned n_per_t = (unsigned)(total / TSTEPS);   // 4,194,304
  const unsigned blocks  = (n_per_t + BLK * 4 - 1) / (BLK * 4);

  if_scan_async<<<blocks, BLK, 0, stream>>>(x, out, n_per_t);
}